// Detect_37847251812632
// MI455X (gfx1250) — compile-verified
//
#include <hip/hip_runtime.h>
#include <cstdint>

// SSD Detect post-processing for MI455X (gfx1250).
// B=128, P=8732 priors, 21 classes, top-200 + NMS(0.45) per (batch, class).
//
// Memory roofline: ~94MB conf + ~8MB scattered loc/prior + ~11MB out
// => ~5us at 23.3 TB/s. One workgroup per (batch, class) = 2560 blocks.
// The per-class score column (stride 84B) is gathered into LDS by the
// Tensor Data Mover: 4 descriptors issued by 4 different waves (TDM ops
// from different waves proceed concurrently). Top-200 is an exact MSD
// radix select on unique 64-bit keys (score-bits | ~idx, matching
// lax.top_k tie-breaking), finished with a 256-wide bitonic sort.

#define NUMC 21
#define TOPK 200
#define NP   8732
#define NB   128
#define CONF_TH 0.01f
#define NMS_TH  0.45f
#define THREADS 256
#define TDM_SPLIT 4
#define TDM_ROWS (NP / TDM_SPLIT)   // 2183 rows per descriptor (exact)

typedef uint32_t u32;
typedef uint64_t u64;
typedef u32 u32x4 __attribute__((ext_vector_type(4)));
typedef int i32x4 __attribute__((ext_vector_type(4)));
typedef int i32x8 __attribute__((ext_vector_type(8)));

__device__ __forceinline__ u64 make_key(float s, int i) {
  u32 b = __float_as_uint(s);
  u32 fk = (b & 0x80000000u) ? ~b : (b | 0x80000000u); // monotone flip
  return ((u64)fk << 32) | (u32)(~(u32)i);             // smaller idx wins ties
}
__device__ __forceinline__ float key_score(u64 k) {
  u32 fk = (u32)(k >> 32);
  u32 b = (fk & 0x80000000u) ? (fk & 0x7FFFFFFFu) : ~fk;
  return __uint_as_float(b);
}

__global__ void zero_out_kernel(float4* __restrict__ out4, int n4) {
  int i = blockIdx.x * blockDim.x + threadIdx.x;
  int stride = gridDim.x * blockDim.x;
  float4 z = make_float4(0.f, 0.f, 0.f, 0.f);
  for (; i < n4; i += stride) out4[i] = z;
}

__global__ __launch_bounds__(THREADS)
void detect_kernel(const float* __restrict__ loc,    // [B, P, 4]
                   const float* __restrict__ conf,   // [B, P, 21]
                   const float* __restrict__ prior,  // [P, 4]
                   float* __restrict__ out)          // [B, 21, 200, 5]
{
  const int cls = blockIdx.x + 1;      // foreground classes 1..20
  const int b   = blockIdx.y;
  const int t   = threadIdx.x;

  __shared__ float s_score[NP];
  __shared__ u32   s_hist[256];
  __shared__ u64   s_cand[256];
  __shared__ float s_x1[TOPK], s_y1[TOPK], s_x2[TOPK], s_y2[TOPK];
  __shared__ float s_sc[TOPK], s_area[TOPK];
  __shared__ int   s_valid[TOPK], s_supp[TOPK], s_keep[TOPK];
  __shared__ int   s_scan[THREADS];
  __shared__ u64   s_prefix;
  __shared__ int   s_need, s_above, s_done, s_cnt;

  // ---- Stage 1: gather conf[b, :, cls] (stride 21 floats) into LDS ----
  // Four TDM descriptors issued by four waves run concurrently in the DMA.
#if __has_builtin(__builtin_amdgcn_tensor_load_to_lds)
  {
    const int wave = t >> 5;
    if (wave < TDM_SPLIT) {
      const int row0 = wave * TDM_ROWS;
      u64 gaddr = (u64)(uintptr_t)(conf +
                   ((size_t)b * NP + (size_t)row0) * NUMC + cls);
      u32 laddr = (u32)(uintptr_t)&s_score[row0]; // low 32 bits of flat = LDS

      u32x4 g0;
      g0.x = 1u;                                  // count=1, user descriptor
      g0.y = laddr;                               // lds_addr
      g0.z = (u32)gaddr;                          // global_addr[31:0]
      g0.w = (u32)((gaddr >> 32) & 0x01FFFFFFu) | (2u << 30); // addr|type=2

      i32x8 g1;
      g1[0] = (int)(2u << 16);            // data_size = 2 (4 bytes)
      g1[1] = (int)((u32)NUMC << 16);     // tensor_dim0 = 21
      g1[2] = (int)((u32)TDM_ROWS << 16); // tensor_dim1 = 2183
      g1[3] = (int)(1u << 16);            // tile_dim0 = 1
      g1[4] = TDM_ROWS;                   // tile_dim1 = 2183 (tile_dim2 = 0)
      g1[5] = NUMC;                       // tensor_dim0_stride = 21 elements
      g1[6] = 0;
      g1[7] = 0;

      i32x4 gz = {0, 0, 0, 0};            // groups 2/3 unused (2D tensor)
#if defined(__clang_major__) && (__clang_major__ >= 23)
      i32x8 gz8 = {0, 0, 0, 0, 0, 0, 0, 0};
      __builtin_amdgcn_tensor_load_to_lds(g0, g1, gz, gz, gz8, 0);
#else
      __builtin_amdgcn_tensor_load_to_lds(g0, g1, gz, gz, 0);
#endif
      __builtin_amdgcn_s_wait_tensorcnt(0);
    }
  }
#else
  for (int i = t; i < NP; i += THREADS)
    s_score[i] = conf[((size_t)b * NP + i) * NUMC + cls];
#endif
  __syncthreads();

  // ---- Stage 2: exact top-200 via MSD radix select on 64-bit keys ----
  u64 prefix = 0;
  int shift = 56;
  int need = TOPK;
  int above = 0;

  for (int level = 0; level < 8; ++level) {
    s_hist[t] = 0;
    __syncthreads();
    for (int i = t; i < NP; i += THREADS) {
      u64 k = make_key(s_score[i], i);
      bool match = (level == 0) || ((k >> (shift + 8)) == prefix);
      if (match) atomicAdd(&s_hist[(u32)(k >> shift) & 255u], 1u);
    }
    __syncthreads();

    // Parallel suffix sum over the 256 bins: cum[t] = sum_{bin>=t} hist[bin]
    s_scan[t] = (int)s_hist[t];
    __syncthreads();
    for (int off = 1; off < THREADS; off <<= 1) {
      int v = (t + off < THREADS) ? s_scan[t + off] : 0;
      __syncthreads();
      s_scan[t] += v;
      __syncthreads();
    }
    // Threshold bin: cum[t] >= need and cum[t+1] < need (unique: cum
    // is non-increasing). run = cum[t+1] = count strictly above bin t.
    {
      int cumT = s_scan[t];
      int run  = (t < THREADS - 1) ? s_scan[t + 1] : 0;
      if (cumT >= need && run < need) {
        s_prefix = (prefix << 8) | (u32)t;
        s_above  = above + run;
        s_need   = need - run;
        s_done   = ((above + cumT) <= 256) ? 1 : 0;
      }
    }
    __syncthreads();
    prefix = s_prefix;
    above  = s_above;
    need   = s_need;
    if (s_done) break;
    shift -= 8;
  }

  // Gather candidates (all keys whose processed digits >= prefix): <=256, >=200
  if (t == 0) s_cnt = 0;
  __syncthreads();
  for (int i = t; i < NP; i += THREADS) {
    u64 k = make_key(s_score[i], i);
    if ((k >> shift) >= prefix) {
      int pos = atomicAdd(&s_cnt, 1);
      if (pos < 256) s_cand[pos] = k;
    }
  }
  __syncthreads();
  if (t >= s_cnt) s_cand[t] = 0;   // pad (sorts last)
  __syncthreads();

  // Bitonic sort 256 keys, descending
  for (u32 ksz = 2; ksz <= 256; ksz <<= 1) {
    for (u32 j = ksz >> 1; j > 0; j >>= 1) {
      u32 ixj = (u32)t ^ j;
      if (ixj > (u32)t) {
        u64 a = s_cand[t], c = s_cand[ixj];
        bool descBlk = (((u32)t & ksz) == 0);
        if (descBlk ? (a < c) : (a > c)) { s_cand[t] = c; s_cand[ixj] = a; }
      }
      __syncthreads();
    }
  }

  // ---- Stage 3: decode only the selected 200 boxes ----
  if (t < TOPK) {
    u64 k = s_cand[t];
    u32 idx = ~((u32)k);
    float scr = key_score(k);
    const float4 l  = ((const float4*)loc)[(size_t)b * NP + idx];
    const float4 pr = ((const float4*)prior)[idx];
    float cx = pr.x + (l.x * 0.1f) * pr.z;
    float cy = pr.y + (l.y * 0.1f) * pr.w;
    float w  = pr.z * expf(l.z * 0.2f);
    float h  = pr.w * expf(l.w * 0.2f);
    float x1 = cx - w * 0.5f;
    float y1 = cy - h * 0.5f;
    float x2 = x1 + w;
    float y2 = y1 + h;
    s_x1[t] = x1; s_y1[t] = y1; s_x2[t] = x2; s_y2[t] = y2;
    s_sc[t] = scr;
    s_area[t] = (x2 - x1) * (y2 - y1);
    s_valid[t] = (scr > CONF_TH) ? 1 : 0;
    s_supp[t] = 0;
  }
  __syncthreads();

  // ---- Stage 4: sequential NMS cascade ----
  for (int i = 0; i < TOPK; ++i) {
    bool active = (s_supp[i] == 0) && (s_valid[i] != 0);
    if (t == i) s_keep[i] = active ? 1 : 0;
    if (active && t < TOPK && t > i) {
      float xx1 = fmaxf(s_x1[i], s_x1[t]);
      float yy1 = fmaxf(s_y1[i], s_y1[t]);
      float xx2 = fminf(s_x2[i], s_x2[t]);
      float yy2 = fminf(s_y2[i], s_y2[t]);
      float inter = fmaxf(xx2 - xx1, 0.0f) * fmaxf(yy2 - yy1, 0.0f);
      float iou = inter / (s_area[t] + s_area[i] - inter);
      if (iou > NMS_TH) s_supp[t] = 1;
    }
    __syncthreads();
  }

  // ---- Stage 5: compact kept rows to the front (stable) and write ----
  s_scan[t] = (t < TOPK) ? s_keep[t] : 0;
  __syncthreads();
  for (int off = 1; off < THREADS; off <<= 1) {
    int v = (t >= off) ? s_scan[t - off] : 0;
    __syncthreads();
    s_scan[t] += v;
    __syncthreads();
  }
  if (t < TOPK && s_keep[t]) {
    int pos = s_scan[t] - 1;  // exclusive rank among kept
    float* o = out + ((((size_t)b * NUMC + cls) * TOPK) + pos) * 5;
    o[0] = s_sc[t];
    o[1] = s_x1[t];
    o[2] = s_y1[t];
    o[3] = s_x2[t];
    o[4] = s_y2[t];
  }
}

extern "C" void kernel_launch(void* const* d_in, const int* in_sizes, int n_in,
                              void* d_out, int out_size, void* d_ws, size_t ws_size,
                              hipStream_t stream) {
  (void)in_sizes; (void)n_in; (void)d_ws; (void)ws_size;
  const float* loc   = (const float*)d_in[0];
  const float* conf  = (const float*)d_in[1];
  const float* prior = (const float*)d_in[2];
  float* out = (float*)d_out;

  // Zero full output (class-0 plane + suppressed/empty rows must be 0).
  // out_size = 128*21*200*5 = 2,688,000 floats, divisible by 4.
  int n4 = out_size >> 2;
  zero_out_kernel<<<1024, THREADS, 0, stream>>>((float4*)out, n4);

  dim3 grid(NUMC - 1, NB);  // (20 classes, 128 batches) = 2560 workgroups
  detect_kernel<<<grid, THREADS, 0, stream>>>(loc, conf, prior, out);
}